// GCNWithAttention_52415780880537
// MI455X (gfx1250) — compile-verified
//
#include <hip/hip_runtime.h>
#include <hip/hip_bf16.h>

// ---------------------------------------------------------------------------
// GCN + low-rank global attention forward for MI455X (gfx1250, wave32).
// f32 everywhere (workload is HBM-bound on edge scatter; ~2.3GB traffic).
// Dense GEMMs use V_WMMA_F32_16X16X4_F32.
// ---------------------------------------------------------------------------

typedef float v2f __attribute__((ext_vector_type(2)));
typedef float v8f __attribute__((ext_vector_type(8)));

#define D_IN   128
#define D_OUT  128
#define KRANK  64

// ---------------------------- small utility kernels ------------------------

__global__ void k_fill1(float* __restrict__ p, int n) {
    int i = blockIdx.x * blockDim.x + threadIdx.x;
    if (i < n) p[i] = 1.0f;
}

__global__ void k_zero(float* __restrict__ p, int n) {
    int i = blockIdx.x * blockDim.x + threadIdx.x;
    if (i < n) p[i] = 0.0f;
}

__global__ void k_deg_count(const int* __restrict__ dst, float* __restrict__ deg, int E) {
    int e = blockIdx.x * blockDim.x + threadIdx.x;
    if (e < E) unsafeAtomicAdd(&deg[dst[e]], 1.0f);
}

__global__ void k_rsqrt(float* __restrict__ p, int n) {
    int i = blockIdx.x * blockDim.x + threadIdx.x;
    if (i < n) p[i] = rsqrtf(p[i]);
}

// xloc = h * dinv^2   (self-loop contribution; also initializes the buffer)
__global__ void k_selfloop(const float* __restrict__ h, const float* __restrict__ dinv,
                           float* __restrict__ xloc, int n) {
    size_t i = (size_t)blockIdx.x * blockDim.x + threadIdx.x;
    if (i >= (size_t)n * D_OUT) return;
    int node = (int)(i >> 7);
    float di = dinv[node];
    xloc[i] = h[i] * di * di;
}

// per-edge scatter: xloc[dst] += h[src] * dinv[src]*dinv[dst]  (2 edges / block)
__global__ void k_edge_scatter(const int* __restrict__ src, const int* __restrict__ dst,
                               const float* __restrict__ h, const float* __restrict__ dinv,
                               float* __restrict__ xloc, int E) {
    int e = blockIdx.x * 2 + (threadIdx.x >> 7);
    if (e >= E) return;
    int d = threadIdx.x & 127;
    int s = src[e], t = dst[e];
    float nrm = dinv[s] * dinv[t];
    unsafeAtomicAdd(&xloc[(size_t)t * D_OUT + d], h[(size_t)s * D_OUT + d] * nrm);
}

__global__ void k_bias_relu(float* __restrict__ xloc, const float* __restrict__ b, int n) {
    size_t i = (size_t)blockIdx.x * blockDim.x + threadIdx.x;
    if (i >= (size_t)n * D_OUT) return;
    xloc[i] = fmaxf(xloc[i] + b[i & 127], 0.0f);
}

// column sums of U (t cols 0..63) and V (t cols 64..127) -> sums[0..63], sums[64..127]
__global__ void k_colsums(const float* __restrict__ t, float* __restrict__ sums, int n) {
    int c = threadIdx.x;            // 64 threads
    float aU = 0.f, aV = 0.f;
    for (int node = blockIdx.x; node < n; node += gridDim.x) {
        aU += t[(size_t)node * 256 + c];
        aV += t[(size_t)node * 256 + 64 + c];
    }
    unsafeAtomicAdd(&sums[c], aU);
    unsafeAtomicAdd(&sums[64 + c], aV);
}

// split-K partial of VtZ (64x64), V = t cols 64..127, Z = t cols 128..191
__global__ void k_vtz(const float* __restrict__ t, float* __restrict__ vtz, int n) {
    __shared__ float sv[2][64];
    __shared__ float sz[2][64];
    int tid = threadIdx.x;          // 256
    int i  = tid >> 2;              // 0..63 : VtZ row
    int j0 = (tid & 3) * 16;        // col group base
    float c[16];
#pragma unroll
    for (int jj = 0; jj < 16; ++jj) c[jj] = 0.f;

    int chunk = (n + gridDim.x - 1) / gridDim.x;
    int n0 = blockIdx.x * chunk;
    int n1 = min(n0 + chunk, n);
    for (int nb = n0; nb < n1; nb += 2) {
        int sub = tid >> 7;         // 0..1 : which node
        int r   = tid & 127;        // 0..127 : which float of [V|Z] row
        int node = nb + sub;
        float val = 0.f;
        if (node < n1) val = t[(size_t)node * 256 + 64 + r];
        if (r < 64) sv[sub][r] = val; else sz[sub][r - 64] = val;
        __syncthreads();
#pragma unroll
        for (int s = 0; s < 2; ++s) {
            float v = sv[s][i];
#pragma unroll
            for (int jj = 0; jj < 16; ++jj) c[jj] += v * sz[s][j0 + jj];
        }
        __syncthreads();
    }
#pragma unroll
    for (int jj = 0; jj < 16; ++jj)
        unsafeAtomicAdd(&vtz[i * 64 + j0 + jj], c[jj]);
}

// Build fused 256x128 weight: rows 0..127 = W_red top, 128..191 = D*VtZ@W_red_mid,
// 192..255 = W_red bottom.  D = n / dot(sumU, sumV).
__global__ void k_build_wcat(const float* __restrict__ W_red, const float* __restrict__ vtz,
                             const float* __restrict__ sums, float* __restrict__ wcat, int n) {
    float dot = 0.f;
#pragma unroll
    for (int k = 0; k < 64; ++k) dot += sums[k] * sums[64 + k];
    float D = (float)n / dot;
    int tid = threadIdx.x;          // 256
    for (int idx = tid; idx < 128 * 128; idx += 256)
        wcat[idx] = W_red[idx];                                   // rows 0..127
    for (int idx = tid; idx < 64 * 128; idx += 256)
        wcat[192 * 128 + idx] = W_red[192 * 128 + idx];           // rows 192..255
    for (int idx = tid; idx < 64 * 128; idx += 256) {             // rows 128..191
        int a = idx >> 7, j = idx & 127;
        float s = 0.f;
#pragma unroll 8
        for (int b = 0; b < 64; ++b)
            s += vtz[a * 64 + b] * W_red[(128 + b) * 128 + j];
        wcat[(128 + a) * 128 + j] = D * s;
    }
}

// ------------------------------ WMMA GEMMs ---------------------------------
// C[M,NCOLS] = A[M,KDIM] @ W[KDIM,NCOLS]  (+bias, relu) using v_wmma_f32_16x16x4_f32.
// 8 waves/block, each wave owns a 16-row strip across all NCOLS/16 tiles.

template <int KDIM, int NCOLS, bool BIAS, bool RELU>
__global__ void k_wmma_gemm(const float* __restrict__ A, const float* __restrict__ W,
                            const float* __restrict__ bias, float* __restrict__ C, int nrows) {
    constexpr int NT = NCOLS / 16;
    int wave = threadIdx.x >> 5, lane = threadIdx.x & 31;
    int mtiles = (nrows + 15) >> 4;
    int mt = blockIdx.x * 8 + wave;
    if (mt >= mtiles) return;                 // wave-uniform exit
    int m0 = mt << 4;
    int am = lane & 15, kh = lane >> 4;       // A: row m0+am, k-pair 2*kh
    int arow = min(m0 + am, nrows - 1);
    const float* Ap = A + (size_t)arow * KDIM + 2 * kh;

    v8f acc[NT] = {};
    for (int k = 0; k < KDIM; k += 4) {
        v2f a = *(const v2f*)(Ap + k);
        const float* W0 = W + (size_t)(k + 2 * kh) * NCOLS + am;
#pragma unroll
        for (int nt = 0; nt < NT; ++nt) {
            v2f b;
            b.x = W0[nt * 16];
            b.y = W0[nt * 16 + NCOLS];
            acc[nt] = __builtin_amdgcn_wmma_f32_16x16x4_f32(
                false, a, false, b, (short)0, acc[nt], false, false);
        }
    }
    int col0 = lane & 15;
#pragma unroll
    for (int nt = 0; nt < NT; ++nt) {
        int col = nt * 16 + col0;
        float bv = BIAS ? bias[col] : 0.f;
#pragma unroll
        for (int j = 0; j < 8; ++j) {
            int row = m0 + j + 8 * kh;
            if (row < nrows) {
                float v = acc[nt][j] + bv;
                if (RELU) v = fmaxf(v, 0.f);
                C[(size_t)row * NCOLS + col] = v;
            }
        }
    }
}

// Final GEMM: A row = [x_local(128) | t cols 0..63 (U) | t cols 192..255 (T)],
// K=256, W=wcat(256x128), +b_red.  Region select is k-uniform per step of 4.
__global__ void k_wmma_gemm_final(const float* __restrict__ xloc, const float* __restrict__ t,
                                  const float* __restrict__ wcat, const float* __restrict__ bias,
                                  float* __restrict__ C, int nrows) {
    constexpr int NT = 8;                     // 128 output cols
    int wave = threadIdx.x >> 5, lane = threadIdx.x & 31;
    int mtiles = (nrows + 15) >> 4;
    int mt = blockIdx.x * 8 + wave;
    if (mt >= mtiles) return;
    int m0 = mt << 4;
    int am = lane & 15, kh = lane >> 4;
    int arow = min(m0 + am, nrows - 1);

    v8f acc[NT] = {};
    for (int k = 0; k < 256; k += 4) {
        const float* Ap; int ld, kc;
        if (k < 128)      { Ap = xloc; ld = 128; kc = k; }
        else if (k < 192) { Ap = t;    ld = 256; kc = k - 128; }  // U
        else              { Ap = t;    ld = 256; kc = k; }        // T (cols 192..255)
        v2f a = *(const v2f*)(Ap + (size_t)arow * ld + kc + 2 * kh);
        const float* W0 = wcat + (size_t)(k + 2 * kh) * 128 + am;
#pragma unroll
        for (int nt = 0; nt < NT; ++nt) {
            v2f b;
            b.x = W0[nt * 16];
            b.y = W0[nt * 16 + 128];
            acc[nt] = __builtin_amdgcn_wmma_f32_16x16x4_f32(
                false, a, false, b, (short)0, acc[nt], false, false);
        }
    }
    int col0 = lane & 15;
#pragma unroll
    for (int nt = 0; nt < NT; ++nt) {
        int col = nt * 16 + col0;
        float bv = bias[col];
#pragma unroll
        for (int j = 0; j < 8; ++j) {
            int row = m0 + j + 8 * kh;
            if (row < nrows)
                C[(size_t)row * 128 + col] = acc[nt][j] + bv;
        }
    }
}

// ------------------------------- launcher ----------------------------------

extern "C" void kernel_launch(void* const* d_in, const int* in_sizes, int n_in,
                              void* d_out, int out_size, void* d_ws, size_t ws_size,
                              hipStream_t stream) {
    const float* x     = (const float*)d_in[0];
    const int*   eidx  = (const int*)d_in[1];
    const float* W_gcn = (const float*)d_in[2];
    const float* b_gcn = (const float*)d_in[3];
    const float* W_att = (const float*)d_in[4];
    const float* b_att = (const float*)d_in[5];
    const float* W_red = (const float*)d_in[6];
    const float* b_red = (const float*)d_in[7];
    float* out = (float*)d_out;

    const int N = in_sizes[0] / D_IN;
    const int E = in_sizes[1] / 2;
    const int* src = eidx;
    const int* dst = eidx + E;

    // workspace layout (floats)
    float* ws   = (float*)d_ws;
    float* h    = ws;                         // N*128
    float* xloc = h    + (size_t)N * 128;     // N*128
    float* tmat = xloc + (size_t)N * 128;     // N*256
    float* dinv = tmat + (size_t)N * 256;     // N
    float* vtz  = dinv + N;                   // 64*64
    float* sums = vtz  + 64 * 64;             // 128
    float* wcat = sums + 128;                 // 256*128
    (void)ws_size; (void)n_in; (void)out_size;

    const int mtiles = (N + 15) / 16;
    const int ggemm  = (mtiles + 7) / 8;
    const int gnodes = (N + 255) / 256;
    const int gelems = (int)(((size_t)N * 128 + 255) / 256);

    // degrees (with self-loop) -> dinv
    k_fill1    <<<gnodes,            256, 0, stream>>>(dinv, N);
    k_deg_count<<<(E + 255) / 256,   256, 0, stream>>>(dst, dinv, E);
    k_rsqrt    <<<gnodes,            256, 0, stream>>>(dinv, N);

    // dense projections (WMMA)
    k_wmma_gemm<128, 128, false, false><<<ggemm, 256, 0, stream>>>(x, W_gcn, nullptr, h, N);
    k_wmma_gemm<128, 256, true,  true ><<<ggemm, 256, 0, stream>>>(x, W_att, b_att, tmat, N);

    // GCN aggregation
    k_selfloop    <<<gelems,          256, 0, stream>>>(h, dinv, xloc, N);
    k_edge_scatter<<<(E + 1) / 2,     256, 0, stream>>>(src, dst, h, dinv, xloc, E);
    k_bias_relu   <<<gelems,          256, 0, stream>>>(xloc, b_gcn, N);

    // LRGA reductions
    k_zero   <<<(4096 + 128 + 255) / 256, 256, 0, stream>>>(vtz, 4096 + 128);
    k_colsums<<<256,  64, 0, stream>>>(tmat, sums, N);
    k_vtz    <<<200, 256, 0, stream>>>(tmat, vtz, N);
    k_build_wcat<<<1, 256, 0, stream>>>(W_red, vtz, sums, wcat, N);

    // fused concat GEMM -> output
    k_wmma_gemm_final<<<ggemm, 256, 0, stream>>>(xloc, tmat, wcat, b_red, out, N);
}